// MeanGaussianExactFlow_45938970198267
// MI455X (gfx1250) — compile-verified
//
#include <hip/hip_runtime.h>

// Problem constants (match reference)
#define Bq 256
#define Nq 1024
#define Dq 128
#define Mq 64

typedef __attribute__((ext_vector_type(2))) float f32x2;
typedef __attribute__((ext_vector_type(8))) float f32x8;

// Workspace layout in floats
#define WS_PHT   0                      // [D][M]    = 8192
#define WS_HPHT  8192                   // [M][M]    = 4096
#define WS_BVEC  12288                  // [B][D]    = 32768
#define WS_A     45056                  // [B][D][D] = 4194304

// ---------------------------------------------------------------------------
// Kernel 1: PHT = V_prior @ H^T  (D x M), HPHT = H @ PHT (M x M). One block.
// ---------------------------------------------------------------------------
__global__ __launch_bounds__(256) void k_prep(const float* __restrict__ Vp,
                                              const float* __restrict__ H,
                                              float* __restrict__ ws) {
  __shared__ float sPHT[Dq * Mq];
  int tid = threadIdx.x;
  for (int idx = tid; idx < Dq * Mq; idx += 256) {
    int d = idx >> 6, m = idx & 63;
    float acc = 0.f;
    for (int e = 0; e < Dq; ++e) acc += Vp[d * Dq + e] * H[m * Dq + e];
    sPHT[idx] = acc;
    ws[WS_PHT + idx] = acc;
  }
  __syncthreads();
  for (int idx = tid; idx < Mq * Mq; idx += 256) {
    int m = idx >> 6, n = idx & 63;
    float acc = 0.f;
    for (int d = 0; d < Dq; ++d) acc += H[m * Dq + d] * sPHT[d * Mq + n];
    ws[WS_HPHT + idx] = acc;
  }
}

// ---------------------------------------------------------------------------
// Kernel 2: per-batch. Invert S = lam*HPHT + sigma^2*I (SPD Gauss-Jordan),
// SH = S^-1 @ H, A = -0.5 * PHT @ SH via f32 WMMA, bias vector via factored
// mat-vecs. One block (8 waves) per batch.
// ---------------------------------------------------------------------------
__global__ __launch_bounds__(256) void k_batch(const float* __restrict__ lam_p,
                                               const float* __restrict__ sigma,
                                               const float* __restrict__ z,
                                               const float* __restrict__ H,
                                               const float* __restrict__ mu,
                                               float* __restrict__ ws) {
  __shared__ float sS[Mq * 65];     // padded stride 65
  __shared__ float sSH[Mq * Dq];
  __shared__ float sPHT[Dq * Mq];
  __shared__ float scol[Mq];
  __shared__ float stmA[Mq];
  __shared__ float stmB[Mq];
  __shared__ float sv1[Dq];
  __shared__ float ssv[Dq];
  __shared__ float spiv;

  int tid = threadIdx.x;
  int b = blockIdx.x;
  float lam = lam_p[0];
  float sg = sigma[b];
  float sig2 = sg * sg;

  const float* gPHT = ws + WS_PHT;
  const float* gHPHT = ws + WS_HPHT;

  for (int idx = tid; idx < Mq * Mq; idx += 256) {
    int i = idx >> 6, j = idx & 63;
    sS[i * 65 + j] = lam * gHPHT[idx] + ((i == j) ? sig2 : 0.f);
  }
  for (int idx = tid; idx < Dq * Mq; idx += 256) sPHT[idx] = gPHT[idx];
  __syncthreads();

  // In-place Gauss-Jordan inversion (SPD => no pivoting).
  for (int k = 0; k < Mq; ++k) {
    if (tid == 0) spiv = 1.0f / sS[k * 65 + k];
    __syncthreads();
    float p = spiv;
    if (tid < Mq) scol[tid] = sS[tid * 65 + k];
    __syncthreads();
    if (tid < Mq) sS[k * 65 + tid] = (tid == k) ? p : sS[k * 65 + tid] * p;
    __syncthreads();
    for (int idx = tid; idx < Mq * Mq; idx += 256) {
      int i = idx >> 6, j = idx & 63;
      if (i != k) {
        if (j == k) sS[i * 65 + k] = -scol[i] * p;
        else        sS[i * 65 + j] -= scol[i] * sS[k * 65 + j];
      }
    }
    __syncthreads();
  }

  // SH = S^-1 @ H : [M][D]
  for (int idx = tid; idx < Mq * Dq; idx += 256) {
    int m = idx >> 7, e = idx & 127;
    float acc = 0.f;
    for (int j = 0; j < Mq; ++j) acc += sS[m * 65 + j] * H[j * Dq + e];
    sSH[idx] = acc;
  }
  __syncthreads();

  // A = -0.5 * PHT @ SH using V_WMMA_F32_16X16X4_F32 (exact fp32).
  float* Ab = ws + WS_A + (size_t)b * Dq * Dq;
  int lane = tid & 31, wv = tid >> 5;
  int r16 = lane & 15, hi = lane >> 4;
  for (int it = 0; it < 8; ++it) {
    int t = it * 8 + wv;             // 64 tiles over 8 waves
    int d0 = (t >> 3) * 16, e0 = (t & 7) * 16;
    f32x8 acc = {};
    for (int m0 = 0; m0 < Mq; m0 += 4) {
      int ka = m0 + 2 * hi;
      f32x2 a, bb;
      a.x = sPHT[(d0 + r16) * Mq + ka];
      a.y = sPHT[(d0 + r16) * Mq + ka + 1];
      bb.x = sSH[ka * Dq + e0 + r16];
      bb.y = sSH[(ka + 1) * Dq + e0 + r16];
      acc = __builtin_amdgcn_wmma_f32_16x16x4_f32(false, a, false, bb,
                                                  (short)0, acc, false, false);
    }
    for (int r = 0; r < 8; ++r) {
      int d = d0 + r + 8 * hi;
      Ab[d * Dq + e0 + r16] = -0.5f * acc[r];
    }
  }
  __syncthreads();

  // Bias vector: v1 = (1/s^2) PHT z ; t1 = (I+lam A)v1 ; t2 = A mu ;
  // s = t1+t2 ; b = s + 2 lam A s.   (A v = -0.5 * PHT @ (SH @ v))
  const float* zb = z + b * Mq;
  float inv_s2 = 1.0f / sig2;
  if (tid < Dq) {
    float acc = 0.f;
    for (int m = 0; m < Mq; ++m) acc += sPHT[tid * Mq + m] * zb[m];
    sv1[tid] = inv_s2 * acc;
  }
  __syncthreads();
  if (tid < Mq) {
    float a0 = 0.f, a1 = 0.f;
    for (int d = 0; d < Dq; ++d) {
      float s = sSH[tid * Dq + d];
      a0 += s * sv1[d];
      a1 += s * mu[d];
    }
    stmA[tid] = a0; stmB[tid] = a1;
  }
  __syncthreads();
  if (tid < Dq) {
    float a0 = 0.f, a1 = 0.f;
    for (int m = 0; m < Mq; ++m) {
      float ph = sPHT[tid * Mq + m];
      a0 += ph * stmA[m];
      a1 += ph * stmB[m];
    }
    float Av1 = -0.5f * a0, Amu = -0.5f * a1;
    ssv[tid] = (sv1[tid] + lam * Av1) + Amu;
  }
  __syncthreads();
  if (tid < Mq) {
    float a0 = 0.f;
    for (int d = 0; d < Dq; ++d) a0 += sSH[tid * Dq + d] * ssv[d];
    stmA[tid] = a0;
  }
  __syncthreads();
  if (tid < Dq) {
    float a0 = 0.f;
    for (int m = 0; m < Mq; ++m) a0 += sPHT[tid * Mq + m] * stmA[m];
    float As = -0.5f * a0;
    ws[WS_BVEC + b * Dq + tid] = ssv[tid] + 2.0f * lam * As;
  }
}

// ---------------------------------------------------------------------------
// Kernel 3: f[b] = x[b] @ A_b^T + bvec_b, the memory-bound workhorse.
// Block = (batch, 128-row n-block), 8 waves x 16-row strips.
// A_b is pre-swizzled into LDS in exact WMMA-B fragment order so the hot-loop
// ds_load_b64 is lane-contiguous (conflict-free across all 64 banks).
// ---------------------------------------------------------------------------
__global__ __launch_bounds__(256) void k_flow(const float* __restrict__ x,
                                              const float* __restrict__ ws,
                                              float* __restrict__ f) {
  __shared__ f32x2 sAf[8192];   // 8 d-tiles * 32 k-steps * 32 lanes = 64 KB
  int tid = threadIdx.x;
  int bb = blockIdx.x >> 3;     // batch
  int nb = blockIdx.x & 7;      // n-block
  const float* A = ws + WS_A + (size_t)bb * Dq * Dq;

  // Swizzle A into per-lane B-fragment order:
  //   B frag (tile d0, step k0): lane l -> { A[d0*16+(l&15)][k0*4+2*(l>>4)], +1 }
  for (int e = tid; e < 8192; e += 256) {
    int d0t = e >> 10;
    int rem = e & 1023;
    int k0 = rem >> 5;
    int ln = rem & 31;
    int row = d0t * 16 + (ln & 15);
    int col = k0 * 4 + 2 * (ln >> 4);
    f32x2 v;
    v.x = A[row * Dq + col];
    v.y = A[row * Dq + col + 1];
    sAf[e] = v;
  }
  __syncthreads();

  int lane = tid & 31, wv = tid >> 5;
  int r16 = lane & 15, hi = lane >> 4;
  int n0 = nb * 128 + wv * 16;
  const float* xs = x + ((size_t)bb * Nq + n0) * Dq;

  // A-operand fragments of x: lane l holds row (l&15), cols k0*4+2*(l>>4)+{0,1}
  int rowoff = r16 * Dq + 2 * hi;
  f32x2 af[32];
#pragma unroll
  for (int k0 = 0; k0 < 32; ++k0)
    af[k0] = *(const f32x2*)(xs + rowoff + k0 * 4);

  // Init accumulators with the bias (folds the +b add into WMMA accumulate).
  const float* bv = ws + WS_BVEC + bb * Dq;
  f32x8 acc[8];
#pragma unroll
  for (int d0 = 0; d0 < 8; ++d0) {
    float bval = bv[d0 * 16 + r16];
#pragma unroll
    for (int r = 0; r < 8; ++r) acc[d0][r] = bval;
  }

  // Main loop: 32 K-steps x 8 d-tiles of v_wmma_f32_16x16x4_f32.
  for (int k0 = 0; k0 < 32; ++k0) {
    f32x2 a = af[k0];
#pragma unroll
    for (int d0 = 0; d0 < 8; ++d0) {
      f32x2 bfr = sAf[(d0 << 10) + (k0 << 5) + lane];
      acc[d0] = __builtin_amdgcn_wmma_f32_16x16x4_f32(false, a, false, bfr,
                                                      (short)0, acc[d0], false, false);
    }
  }

  // Store: C layout -> lane l, vgpr r holds f[n0 + r + 8*(l>>4)][d0*16+(l&15)]
  float* fb = f + ((size_t)bb * Nq + n0) * Dq;
#pragma unroll
  for (int d0 = 0; d0 < 8; ++d0) {
#pragma unroll
    for (int r = 0; r < 8; ++r)
      fb[(r + 8 * hi) * Dq + d0 * 16 + r16] = acc[d0][r];
  }
}

// ---------------------------------------------------------------------------
extern "C" void kernel_launch(void* const* d_in, const int* in_sizes, int n_in,
                              void* d_out, int out_size, void* d_ws, size_t ws_size,
                              hipStream_t stream) {
  (void)in_sizes; (void)n_in; (void)out_size; (void)ws_size;
  const float* lam   = (const float*)d_in[0];
  const float* x     = (const float*)d_in[1];
  const float* H     = (const float*)d_in[2];
  const float* sigma = (const float*)d_in[3];
  const float* z     = (const float*)d_in[4];
  const float* Vp    = (const float*)d_in[5];
  const float* mu    = (const float*)d_in[6];
  float* f  = (float*)d_out;
  float* ws = (float*)d_ws;

  hipLaunchKernelGGL(k_prep,  dim3(1),      dim3(256), 0, stream, Vp, H, ws);
  hipLaunchKernelGGL(k_batch, dim3(Bq),     dim3(256), 0, stream, lam, sigma, z, H, mu, ws);
  hipLaunchKernelGGL(k_flow,  dim3(Bq * 8), dim3(256), 0, stream, x, ws, f);
}